// Corr_Layer_89146341196399
// MI455X (gfx1250) — compile-verified
//
#include <hip/hip_runtime.h>

typedef float v2f __attribute__((ext_vector_type(2)));
typedef float v8f __attribute__((ext_vector_type(8)));

#define B_   256
#define C_   8
#define F_   8
#define T_   4096
#define K_   9      // conv kernel height
#define CF   64     // C_*F_ flattened rows of x per batch
#define OUTC 9
#define ROWS_OUT 72 // OUTC*F_ flattened rows of out per batch

// One wave32 per 16-wide t tile: fused (x -> out copy) + WMMA f32 16x16x4
// chain (K=64 as 16 steps) producing corr for channel 8, + bias.
__global__ __launch_bounds__(256) void Corr_Layer_89146341196399_kernel(
    const float* __restrict__ x, const float* __restrict__ W,
    const float* __restrict__ bias, float* __restrict__ out)
{
  const int tid  = threadIdx.x;
  const int lane = tid & 31;
  const int wave = tid >> 5;
  const int m    = lane & 15;   // column within tile / A-row (M)
  const int hi   = lane >> 4;   // lane-half selects K offset (+2)

  const int bidx  = blockIdx.x >> 5;   // batch
  const int tchnk = blockIdx.x & 31;   // 32 chunks of 128 t each
  const int t0    = tchnk * 128 + wave * 16;

  const float b0 = bias[0];

  // Combined weight matrix A[f, c*8+f'] = W[c, f'+1] + (f'==f) * W[c, 0].
  // 16x64 (rows 8..15 zero), stored as 16 K=4 fragments per ISA A-layout:
  //   VGPR0: K=4kk+0 (lanes 0-15), K=4kk+2 (lanes 16-31); VGPR1: +1.
  v2f afrag[16];
#pragma unroll
  for (int kk = 0; kk < 16; ++kk) {
    const int k0 = kk * 4 + hi * 2;
    const int c0 = k0 >> 3, f0 = k0 & 7;
    const int k1 = k0 + 1;
    const int c1 = k1 >> 3, f1 = k1 & 7;
    float a0 = 0.f, a1 = 0.f;
    if (m < 8) {
      a0 = W[c0 * K_ + f0 + 1] + (f0 == m ? W[c0 * K_] : 0.f);
      a1 = W[c1 * K_ + f1 + 1] + (f1 == m ? W[c1 * K_] : 0.f);
    }
    afrag[kk].x = a0;
    afrag[kk].y = a1;
  }

  v8f acc;
#pragma unroll
  for (int i = 0; i < 8; ++i) acc[i] = b0;   // bias folded into accumulator

  const size_t xbase = (size_t)bidx * CF * T_ + (size_t)t0 + m;
  const size_t obase = (size_t)bidx * ROWS_OUT * T_ + (size_t)t0 + m;

  // K=64 reduction: 16 chained f32 WMMAs. Each B-fragment load is also the
  // x -> out channel copy (out rows 0..63 == x rows), so x is read once.
#pragma unroll
  for (int kk = 0; kk < 16; ++kk) {
    const int row0 = kk * 4 + hi * 2;     // B-layout: K=row0 in this half
    const int row1 = row0 + 1;
    const float v0 = __builtin_nontemporal_load(&x[xbase + (size_t)row0 * T_]);
    const float v1 = __builtin_nontemporal_load(&x[xbase + (size_t)row1 * T_]);
    __builtin_nontemporal_store(v0, &out[obase + (size_t)row0 * T_]);
    __builtin_nontemporal_store(v1, &out[obase + (size_t)row1 * T_]);
    v2f bfrag;
    bfrag.x = v0;
    bfrag.y = v1;
    // 8 args: (neg_a, A, neg_b, B, c_mod, C, reuse_a, reuse_b)
    acc = __builtin_amdgcn_wmma_f32_16x16x4_f32(
        false, afrag[kk], false, bfrag, (short)0, acc, false, false);
  }

  // D layout: VGPR r, lanes 0-15 -> (M=r, N=lane). Only f=0..7 valid.
  if (hi == 0) {
#pragma unroll
    for (int r = 0; r < 8; ++r) {
      __builtin_nontemporal_store(acc[r], &out[obase + (size_t)(64 + r) * T_]);
    }
  }
}

extern "C" void kernel_launch(void* const* d_in, const int* in_sizes, int n_in,
                              void* d_out, int out_size, void* d_ws, size_t ws_size,
                              hipStream_t stream) {
  const float* x  = (const float*)d_in[0];   // [256, 8, 8, 4096] f32
  const float* W  = (const float*)d_in[1];   // [8, 9] f32
  const float* bb = (const float*)d_in[2];   // [1] f32
  float* out      = (float*)d_out;           // [256, 9, 8, 4096] f32

  // 256 batches * 32 t-chunks (128 t each) ; 256 threads = 8 waves,
  // each wave owns a 16-wide t tile.
  dim3 grid(B_ * (T_ / 128));
  dim3 block(256);
  hipLaunchKernelGGL(Corr_Layer_89146341196399_kernel, grid, block, 0, stream,
                     x, W, bb, out);
}